// DAMEDMedian_82102594830596
// MI455X (gfx1250) — compile-verified
//
#include <hip/hip_runtime.h>
#include <math.h>

typedef float v2f __attribute__((ext_vector_type(2)));
typedef float v8f __attribute__((ext_vector_type(8)));

// Branchless f32 erf — the Eigen/XLA "generic_fast_erf_float" rational that
// jax.lax.erf lowers to for f32 on GPU: clamp to [-4,4], x*P6(x^2)/Q4(x^2).
// Straight-line: 10 FMA + 3 MUL + clamp + v_rcp_f32. No EXEC divergence.
__device__ __forceinline__ float erf_xla(float x) {
    x = fminf(4.0f, fmaxf(-4.0f, x));
    const float x2 = x * x;
    float p = fmaf(x2, -2.72614225801306e-10f, 2.77068142495902e-08f);
    p = fmaf(x2, p, -2.10102402082508e-06f);
    p = fmaf(x2, p, -5.69250639462346e-05f);
    p = fmaf(x2, p, -7.34990630326855e-04f);
    p = fmaf(x2, p, -2.95459980854025e-03f);
    p = fmaf(x2, p, -1.60960333262415e-02f);
    p = x * p;
    float q = fmaf(x2, -1.45660718464996e-05f, -2.13374055278905e-04f);
    q = fmaf(x2, q, -1.68282697438203e-03f);
    q = fmaf(x2, q, -7.37332916720468e-03f);
    q = fmaf(x2, q, -1.42647390514189e-02f);
    return p * __builtin_amdgcn_rcpf(q);   // v_rcp_f32: ~1 ulp, plenty for fm sign
}

// Wave32: each wave owns 16 coordinates. Lane l -> coordinate (l&15); the two
// half-waves split the 64 workers (32 register-resident values per lane).
// Worker-sum reduction: 16 chained V_WMMA_F32_16X16X4_F32 with A = ones
// (layout-invariant in B; fm[lane&15] lands in acc[0] of every lane).
__global__ __launch_bounds__(256)
void damed_median_wmma_bisect(const float* __restrict__ y,
                              float* __restrict__ out, int D) {
    const int lane = threadIdx.x & 31;
    const int wave = threadIdx.x >> 5;
    const int n    = lane & 15;   // coordinate slot within wave tile
    const int half = lane >> 4;   // worker-set selector

    const size_t dReal = (size_t)blockIdx.x * 128u + (size_t)wave * 16u + (size_t)n;
    size_t d = dReal;
    if (d >= (size_t)D) d = (size_t)D - 1;  // clamp: keep EXEC all-ones for WMMA

    // Load 32 worker values per lane, once, into registers.
    // chunk c (0..15), j (0..1) -> worker w = 4c + 2*half + j  (bijection onto 0..63)
    float v[32];
#pragma unroll
    for (int c = 0; c < 16; ++c) {
#pragma unroll
        for (int j = 0; j < 2; ++j) {
            const int w = c * 4 + half * 2 + j;
            v[c * 2 + j] = y[(size_t)w * (size_t)D + d];
        }
    }

    // Bracket = min/max over all 64 workers (combine half-waves via lane^16).
    float lo = v[0], hi = v[0];
#pragma unroll
    for (int i = 1; i < 32; ++i) { lo = fminf(lo, v[i]); hi = fmaxf(hi, v[i]); }
    lo = fminf(lo, __shfl_xor(lo, 16));
    hi = fmaxf(hi, __shfl_xor(hi, 16));

    const v2f ones = {1.0f, 1.0f};  // A matrix: 16x4 of 1.0
    float xl = lo, xg = hi;

    for (int it = 0; it < 64; ++it) {
        const float xm = 0.5f * (xl + xg);

        v8f acc = {0.f, 0.f, 0.f, 0.f, 0.f, 0.f, 0.f, 0.f};
#pragma unroll
        for (int c = 0; c < 16; ++c) {
            v2f b;
            b.x = erf_xla(v[2 * c]     - xm);   // tau = 1
            b.y = erf_xla(v[2 * c + 1] - xm);
            // fm[n] += sum over this K=4 worker chunk, accumulated in fp32.
            acc = __builtin_amdgcn_wmma_f32_16x16x4_f32(
                /*neg_a=*/false, ones, /*neg_b=*/false, b,
                /*c_mod=*/(short)0, acc, /*reuse_a=*/false, /*reuse_b=*/false);
        }
        const float fm  = acc[0];        // all D rows equal; lane's coord is N=lane&15
        const bool left = fm > 0.0f;
        xl = left ? xm : xl;
        xg = left ? xg : xm;

        // Wave-uniform early exit: once the midpoint is a fixed point in every
        // lane, remaining reference iterations change the output by <= 1 ulp.
        const float nxt = 0.5f * (xl + xg);
        const bool notdone = (nxt != xl) && (nxt != xg);
        if (!__any((int)notdone)) break;  // uniform SCC branch; EXEC untouched
    }

    if (half == 0 && dReal < (size_t)D)
        out[dReal] = 0.5f * (xl + xg);
}

extern "C" void kernel_launch(void* const* d_in, const int* in_sizes, int n_in,
                              void* d_out, int out_size, void* d_ws, size_t ws_size,
                              hipStream_t stream) {
    (void)n_in; (void)d_ws; (void)ws_size; (void)out_size;
    const float* y = (const float*)d_in[0];
    float* out = (float*)d_out;
    const int NW = 64;
    const int D = in_sizes[0] / NW;           // 2^21 for the reference setup
    const int coordsPerBlock = 128;           // 8 waves x 16 coords
    const int grid = (D + coordsPerBlock - 1) / coordsPerBlock;
    damed_median_wmma_bisect<<<grid, 256, 0, stream>>>(y, out, D);
}